// PromptGeneratorDeep_81870666596640
// MI455X (gfx1250) — compile-verified
//
#include <hip/hip_runtime.h>
#include <hip/hip_bf16.h>

// ---------------- problem constants ----------------
constexpr int B = 2, N = 2048, C = 384, K = 20;
constexpr float EPSV = 1e-5f, SLOPE = 0.2f;
constexpr int C2 = 2 * C;            // 768
constexpr int C3 = 3 * C;            // 1152
constexpr int BNC = B * N * C;       // 1,572,864

typedef float v2f __attribute__((ext_vector_type(2)));
typedef float v8f __attribute__((ext_vector_type(8)));

// fp32 WMMA: D(16x16,f32) = A(16x4,f32) * B(4x16,f32) + C
__device__ __forceinline__ v8f wmma4(v2f a, v2f b, v8f c) {
  return __builtin_amdgcn_wmma_f32_16x16x4_f32(false, a, false, b, (short)0, c,
                                               false, false);
}

__device__ __forceinline__ float negInf() { return -__builtin_inff(); }

// ---------------- transpose x: (B,C,N) -> (B,N,C) ----------------
__global__ void transpose_k(const float* __restrict__ x, float* __restrict__ X0) {
  int i = blockIdx.x * 256 + threadIdx.x;
  if (i < BNC) {
    int b = i / (N * C);
    int r = i % (N * C);
    int n = r / C, c = r % C;
    X0[i] = x[((size_t)b * C + c) * N + n];
  }
}

// ---------------- per-row squared norms ----------------
__global__ void rowsq_k(const float* __restrict__ X, float* __restrict__ sq) {
  int lane = threadIdx.x & 31;
  int row = blockIdx.x * 8 + (threadIdx.x >> 5);   // [0, B*N)
  const float* p = X + (size_t)row * C;
  float s = 0.f;
  for (int j = lane; j < C; j += 32) { float v = p[j]; s += v * v; }
  for (int off = 16; off; off >>= 1) s += __shfl_xor(s, off, 32);
  if (lane == 0) sq[row] = s;
}

// ---------------- neg_dist = 2*X·Xt - sq_r - sq_c  (WMMA fp32) ----------------
__global__ void dist_k(const float* __restrict__ X, const float* __restrict__ sq,
                       float* __restrict__ dist) {
  int gw = blockIdx.x * 8 + (threadIdx.x >> 5);    // global wave == tile id
  int lane = threadIdx.x & 31, lm = lane & 15, lh = lane >> 4;
  int tpb = (N / 16) * (N / 16);                   // 16384 tiles per batch
  int b = gw / tpb, t = gw % tpb;
  int rt = t >> 7, ct = t & 127;
  const float* Xb = X + (size_t)b * N * C;
  const float* arow = Xb + (size_t)(rt * 16 + lm) * C + 2 * lh;
  const float* brow = Xb + (size_t)(ct * 16 + lm) * C + 2 * lh;
  v8f acc = {0.f, 0.f, 0.f, 0.f, 0.f, 0.f, 0.f, 0.f};
#pragma unroll 8
  for (int k0 = 0; k0 < C; k0 += 4) {
    v2f a = *(const v2f*)(arow + k0);
    v2f bb = *(const v2f*)(brow + k0);
    acc = wmma4(a, bb, acc);
  }
  int col = ct * 16 + lm;
  float sqc = sq[b * N + col];
  float* dr = dist + (size_t)b * N * N;
#pragma unroll
  for (int r = 0; r < 8; r++) {
    int row = rt * 16 + lh * 8 + r;
    dr[(size_t)row * N + col] = 2.f * acc[r] - sq[b * N + row] - sqc;
  }
}

// ---------------- top-K (K=20) per row, one wave per row ----------------
__global__ void topk_k(const float* __restrict__ dist, int* __restrict__ idxb) {
  int lane = threadIdx.x & 31;
  int row = blockIdx.x * 8 + (threadIdx.x >> 5);   // [0, B*N)
  const float* p = dist + (size_t)row * N;
  float v[64];
#pragma unroll
  for (int j = 0; j < 64; j++) v[j] = p[lane + 32 * j];
  int* op = idxb + row * K;
  for (int t = 0; t < K; t++) {
    float bv = negInf(); int bi = 0x7fffffff;
#pragma unroll
    for (int j = 0; j < 64; j++) {
      float val = v[j]; int ci = lane + 32 * j;
      if (val > bv || (val == bv && ci < bi)) { bv = val; bi = ci; }
    }
    for (int off = 16; off; off >>= 1) {
      float ov = __shfl_xor(bv, off, 32);
      int   oi = __shfl_xor(bi, off, 32);
      if (ov > bv || (ov == bv && oi < bi)) { bv = ov; bi = oi; }
    }
    if (lane == (bi & 31)) {
      int jj = bi >> 5;
#pragma unroll
      for (int j = 0; j < 64; j++) if (j == jj) v[j] = negInf();
    }
    if (lane == 0) op[t] = bi;
  }
}

// ---------------- Wd[o][c] = We[o][C+c] - We[o][c] ----------------
__global__ void wd_k(const float* __restrict__ We, float* __restrict__ Wd) {
  int i = blockIdx.x * 256 + threadIdx.x;
  if (i < C * C) { int o = i / C, c = i % C; Wd[i] = We[o * C2 + C + c] - We[o * C2 + c]; }
}

// ---------------- T = X @ Wd^T  (B,N,C) ----------------
__global__ void tgemm_k(const float* __restrict__ X, const float* __restrict__ Wd,
                        float* __restrict__ T) {
  int gw = blockIdx.x * 8 + (threadIdx.x >> 5);
  int lane = threadIdx.x & 31, lm = lane & 15, lh = lane >> 4;
  int tpb = (N / 16) * (C / 16);                    // 3072 tiles per batch
  int b = gw / tpb, t = gw % tpb;
  int rt = t / (C / 16), ct = t % (C / 16);
  const float* arow = X + ((size_t)b * N + rt * 16 + lm) * C + 2 * lh;
  const float* brow = Wd + (size_t)(ct * 16 + lm) * C + 2 * lh;
  v8f acc = {0.f, 0.f, 0.f, 0.f, 0.f, 0.f, 0.f, 0.f};
#pragma unroll 8
  for (int k0 = 0; k0 < C; k0 += 4) {
    v2f a = *(const v2f*)(arow + k0);
    v2f bb = *(const v2f*)(brow + k0);
    acc = wmma4(a, bb, acc);
  }
  int col = ct * 16 + lm;
  float* Tb = T + (size_t)b * N * C;
#pragma unroll
  for (int r = 0; r < 8; r++) Tb[(size_t)(rt * 16 + lh * 8 + r) * C + col] = acc[r];
}

// ---------------- fused edge-conv: H = Neigh·W1^T + T, BN, LeakyReLU, max_k ----
// one block = 4 points (80 neighbor rows = 5 M-tiles); 8 waves x 3 col-tiles
constexpr int LROW = 388;  // padded LDS row stride (floats) -> conflict-free
constexpr int EDGE_SHMEM = (80 * LROW + 4 * C + C + C + 80) * 4;

__global__ void edge_k(const float* __restrict__ X, const int* __restrict__ idxb,
                       const float* __restrict__ T, const float* __restrict__ We,
                       const float* __restrict__ g, const float* __restrict__ bt,
                       const float* __restrict__ m, const float* __restrict__ vv,
                       float* __restrict__ out) {
  extern __shared__ float sh[];
  float* sh_neigh = sh;                       // 80 x 388
  float* sh_T     = sh_neigh + 80 * LROW;     // 4 x 384
  float* sh_scale = sh_T + 4 * C;             // 384
  float* sh_bias  = sh_scale + C;             // 384
  int*   sh_nbr   = (int*)(sh_bias + C);      // 80

  int tid = threadIdx.x;
  int b = blockIdx.x / (N / 4);
  int pbase = (blockIdx.x % (N / 4)) * 4;

  for (int c = tid; c < C; c += 256) {
    float sc = g[c] * rsqrtf(vv[c] + EPSV);
    sh_scale[c] = sc;
    sh_bias[c]  = bt[c] - m[c] * sc;
  }
  if (tid < 80) {
    int p = tid / K, k = tid % K;
    sh_nbr[tid] = idxb[((size_t)b * N + pbase + p) * K + k];
  }
  __syncthreads();

  const float* Xb = X + (size_t)b * N * C;
  for (int e = tid; e < 80 * C; e += 256) {
    int r = e / C, c = e % C;
    sh_neigh[r * LROW + c] = Xb[(size_t)sh_nbr[r] * C + c];
  }
  const float* Tb = T + ((size_t)b * N + pbase) * C;
  for (int e = tid; e < 4 * C; e += 256) sh_T[e] = Tb[e];
  __syncthreads();

  int wid = tid >> 5, lane = tid & 31, lm = lane & 15, lh = lane >> 4;
  for (int nt = wid * 3; nt < wid * 3 + 3; nt++) {
    int col = nt * 16 + lm;
    const float* bcol = We + (size_t)col * C2 + 2 * lh;   // W1 = We[:, :C]
    float pmax[4] = {negInf(), negInf(), negInf(), negInf()};
    for (int mt = 0; mt < 5; mt++) {
      const float* arow = sh_neigh + (mt * 16 + lm) * LROW + 2 * lh;
      v8f acc = {0.f, 0.f, 0.f, 0.f, 0.f, 0.f, 0.f, 0.f};
#pragma unroll 8
      for (int k0 = 0; k0 < C; k0 += 4) {
        v2f a = *(const v2f*)(arow + k0);
        v2f bb = *(const v2f*)(bcol + k0);
        acc = wmma4(a, bb, acc);
      }
#pragma unroll
      for (int r = 0; r < 8; r++) {
        int row = mt * 16 + lh * 8 + r;
        unsigned p = (unsigned)row / (unsigned)K;          // 0..3
        float h = acc[r] + sh_T[p * C + col];
        h = h * sh_scale[col] + sh_bias[col];
        h = h > 0.f ? h : SLOPE * h;
        pmax[p] = fmaxf(pmax[p], h);
      }
    }
#pragma unroll
    for (int p = 0; p < 4; p++)
      pmax[p] = fmaxf(pmax[p], __shfl_xor(pmax[p], 16, 32));
    if (lh == 0) {
      float* ob = out + ((size_t)b * N + pbase) * C;
#pragma unroll
      for (int p = 0; p < 4; p++) ob[(size_t)p * C + col] = pmax[p];
    }
  }
}

// ---------------- final projection + BN + LeakyReLU + global max pool --------
__device__ __forceinline__ void atomicMaxF(float* addr, float val) {
  unsigned* ua = (unsigned*)addr;
  unsigned old = *ua;
  while (true) {
    float f = __uint_as_float(old);
    if (f >= val) break;
    unsigned assumed = old;
    old = atomicCAS(ua, assumed, __float_as_uint(val));
    if (old == assumed) break;
  }
}

__global__ void initout_k(float* out) {
  int i = blockIdx.x * 256 + threadIdx.x;
  if (i < B * C) out[i] = -__builtin_inff();
}

__global__ void proj_k(const float* __restrict__ X1, const float* __restrict__ X2,
                       const float* __restrict__ X3, const float* __restrict__ Wp,
                       const float* __restrict__ g, const float* __restrict__ bt,
                       const float* __restrict__ m, const float* __restrict__ vv,
                       float* __restrict__ out) {
  int gw = blockIdx.x * 8 + (threadIdx.x >> 5);
  int lane = threadIdx.x & 31, lm = lane & 15, lh = lane >> 4;
  int tpb = (N / 16) * (C / 16);
  int b = gw / tpb, t = gw % tpb;
  int rt = t / (C / 16), ct = t % (C / 16);
  size_t roff = ((size_t)b * N + rt * 16 + lm) * C;
  const float* segs[3] = {X1 + roff, X2 + roff, X3 + roff};
  const float* brow = Wp + (size_t)(ct * 16 + lm) * C3 + 2 * lh;
  v8f acc = {0.f, 0.f, 0.f, 0.f, 0.f, 0.f, 0.f, 0.f};
#pragma unroll 4
  for (int k0 = 0; k0 < C3; k0 += 4) {
    int seg = k0 / C;
    v2f a = *(const v2f*)(segs[seg] + (k0 - seg * C) + 2 * lh);
    v2f bb = *(const v2f*)(brow + k0);
    acc = wmma4(a, bb, acc);
  }
  int col = ct * 16 + lm;
  float sc = g[col] * rsqrtf(vv[col] + EPSV);
  float bi = bt[col] - m[col] * sc;
  float pm = negInf();
#pragma unroll
  for (int r = 0; r < 8; r++) {
    float h = acc[r] * sc + bi;
    h = h > 0.f ? h : SLOPE * h;
    pm = fmaxf(pm, h);
  }
  pm = fmaxf(pm, __shfl_xor(pm, 16, 32));
  if (lh == 0) atomicMaxF(&out[b * C + col], pm);
}

// ---------------- host driver ----------------
extern "C" void kernel_launch(void* const* d_in, const int* in_sizes, int n_in,
                              void* d_out, int out_size, void* d_ws, size_t ws_size,
                              hipStream_t stream) {
  const float* x = (const float*)d_in[0];
  const float* We[3] = {(const float*)d_in[2], (const float*)d_in[7], (const float*)d_in[12]};
  const float* g[3]  = {(const float*)d_in[3], (const float*)d_in[8], (const float*)d_in[13]};
  const float* bt[3] = {(const float*)d_in[4], (const float*)d_in[9], (const float*)d_in[14]};
  const float* mm[3] = {(const float*)d_in[5], (const float*)d_in[10], (const float*)d_in[15]};
  const float* vv[3] = {(const float*)d_in[6], (const float*)d_in[11], (const float*)d_in[16]};
  const float* Wp = (const float*)d_in[17];
  const float* gp = (const float*)d_in[18];
  const float* bp = (const float*)d_in[19];
  const float* mp = (const float*)d_in[20];
  const float* vp = (const float*)d_in[21];
  float* out = (float*)d_out;

  float* ws = (float*)d_ws;
  float* X0 = ws;
  float* X1 = X0 + BNC;
  float* X2 = X1 + BNC;
  float* X3 = X2 + BNC;
  float* T  = X3 + BNC;
  float* Wd = T + BNC;
  float* sq = Wd + C * C;
  float* dist = sq + B * N;
  int* idx = (int*)(dist + (size_t)B * N * N);
  float* Xs[4] = {X0, X1, X2, X3};

  transpose_k<<<(BNC + 255) / 256, 256, 0, stream>>>(x, X0);

  for (int L = 0; L < 3; L++) {
    rowsq_k<<<B * N / 8, 256, 0, stream>>>(Xs[L], sq);
    dist_k<<<B * (N / 16) * (N / 16) / 8, 256, 0, stream>>>(Xs[L], sq, dist);
    topk_k<<<B * N / 8, 256, 0, stream>>>(dist, idx);
    wd_k<<<(C * C + 255) / 256, 256, 0, stream>>>(We[L], Wd);
    tgemm_k<<<B * (N / 16) * (C / 16) / 8, 256, 0, stream>>>(Xs[L], Wd, T);
    edge_k<<<B * (N / 4), 256, EDGE_SHMEM, stream>>>(Xs[L], idx, T, We[L],
                                                     g[L], bt[L], mm[L], vv[L], Xs[L + 1]);
  }

  initout_k<<<(B * C + 255) / 256, 256, 0, stream>>>(out);
  proj_k<<<B * (N / 16) * (C / 16) / 8, 256, 0, stream>>>(X1, X2, X3, Wp,
                                                          gp, bp, mp, vp, out);
}